// SelectiveSSM2_37237366456745
// MI455X (gfx1250) — compile-verified
//
#include <hip/hip_runtime.h>
#include <hip/hip_bf16.h>

// Problem constants (B, L, D, N, R from the reference)
static constexpr int BB = 2;
static constexpr int LL = 1024;
static constexpr int DD = 1024;
static constexpr int NN = 64;
static constexpr int RR = 64;
static constexpr int KX = RR + 2 * NN;   // 192 rows of Wx

static constexpr int KC   = 64;   // K-chunk staged in LDS per iteration
static constexpr int LDSW = 66;   // padded LDS row stride (64 + 2) -> conflict-free ds_load_b64

typedef float v2f __attribute__((ext_vector_type(2)));
typedef float v8f __attribute__((ext_vector_type(8)));
typedef unsigned int uint32x4 __attribute__((ext_vector_type(4)));
typedef int int32x4 __attribute__((ext_vector_type(4)));
typedef int int32x8 __attribute__((ext_vector_type(8)));
typedef __attribute__((address_space(3))) void lds_void;

#if __has_builtin(__builtin_amdgcn_tensor_load_to_lds)
#define HAVE_TDM 1
#if __has_include(<hip/amd_detail/amd_gfx1250_TDM.h>)
#define TDM_SIX_ARGS 1   // therock-10.0 headers -> 6-arg builtin
#endif
#endif

__device__ __forceinline__ float sigm(float v)   { return 1.f / (1.f + __expf(-v)); }
__device__ __forceinline__ float softpl(float v) { return (v > 20.f) ? v : log1pf(__expf(v)); }

// ---------------------------------------------------------------------------
// RMSNorm: one block per (b,l) row of D=1024. Wave32 butterfly + LDS combine.
// ---------------------------------------------------------------------------
__global__ __launch_bounds__(256) void rmsnorm_kernel(const float* __restrict__ x,
                                                      const float* __restrict__ w,
                                                      float* __restrict__ xn) {
    const int row = blockIdx.x;                  // 0 .. B*L-1
    const float* xr = x + (size_t)row * DD;
    float s = 0.f;
    for (int j = threadIdx.x; j < DD; j += 256) { float v = xr[j]; s += v * v; }
    #pragma unroll
    for (int off = 16; off >= 1; off >>= 1) s += __shfl_xor(s, off, 32);
    __shared__ float ps[8];
    const int wid = threadIdx.x >> 5, lane = threadIdx.x & 31;
    if (lane == 0) ps[wid] = s;
    __syncthreads();
    float tot = 0.f;
    #pragma unroll
    for (int i = 0; i < 8; i++) tot += ps[i];
    const float scale = rsqrtf(tot * (1.f / DD) + 1e-6f);
    float* xo = xn + (size_t)row * DD;
    for (int j = threadIdx.x; j < DD; j += 256) xo[j] = xr[j] * scale * w[j];
}

// ---------------------------------------------------------------------------
// fp32 WMMA GEMM:  Out[m,n] = sum_k A[m,k] * W[n,k]
//   Block = 256 threads = 8 waves; block tile = 128(M) x 64(N).
//   Wave computes a 16x64 strip: 1 A fragment reused across 4 WMMA tiles.
//   W chunk (64 rows x 64 k = 16 KB) staged in LDS by the Tensor Data Mover
//   (one TENSOR_LOAD_TO_LDS per chunk, wave 0 only, s_wait_tensorcnt),
//   with D# padding (2 DWORDs per 64) giving a 66-float LDS row stride.
// ---------------------------------------------------------------------------
enum { EP_NONE = 0, EP_SEL = 1, EP_IMP = 2, EP_DELTA = 3, EP_GATE = 4 };

template <int EP>
__global__ __launch_bounds__(256) void gemm_wmma(
    const float* __restrict__ A, int lda,
    const float* __restrict__ W, int ldw, int K,
    float* __restrict__ Out, int ldo,
    const float* __restrict__ bias,    // per-col bias (selb/impb/dtb/gateb)
    const float* __restrict__ gatev,   // sel_gate for EP_SEL
    const float* __restrict__ Xin,     // elementwise companion (xn0/xn1)
    float* __restrict__ Xout,          // gated output (xn1/xn2)
    const float* __restrict__ aux,     // xp (stride KX) for EP_IMP
    float* __restrict__ BsOut,         // B_ssm*imp for EP_IMP
    float* __restrict__ CsOut)         // C_ssm*imp for EP_IMP
{
    __shared__ float lw[64 * LDSW];          // staged W chunk, padded rows

    const int wid  = threadIdx.x >> 5;       // wave in block (0..7)
    const int lane = threadIdx.x & 31;       // wave32 lane
    const int lh   = lane & 15;
    const int hi   = lane >> 4;              // half-wave selector
    const int m0   = blockIdx.x * 128 + wid * 16;
    const int n0b  = blockIdx.y * 64;

    // ISA A layout (16x4 f32): lanes 0-15 hold K={0,1}, lanes 16-31 hold K={2,3}
    const float* Ap = A + (size_t)(m0 + lh) * lda + 2 * hi;

    v8f acc[4] = {};
    for (int kc = 0; kc < K; kc += KC) {
        // ---- stage W[n0b .. n0b+63][kc .. kc+63] into LDS -------------------
#ifdef HAVE_TDM
        if (threadIdx.x < 32) {              // wave 0 issues the TDM transfer
            const uint64_t ga = (uint64_t)(uintptr_t)(W + (size_t)n0b * ldw + kc);
            const uint32_t lo = (uint32_t)(uintptr_t)(lds_void*)lw;   // LDS byte offset
            uint32x4 g0;
            g0.x = 1u;                                      // count=1 (valid D#)
            g0.y = lo;                                      // lds_addr
            g0.z = (uint32_t)ga;                            // global_addr[31:0]
            g0.w = (uint32_t)((ga >> 32) & 0x1FFFFFFu) | (2u << 30);  // addr[56:32]|type=2
            const uint32_t TD = 1u << 30;                   // huge tensor dims (no OOB)
            int32x8 g1;
            g1[0] = (int)((2u << 16)    |   // data_size = 4 bytes
                          (1u << 20)    |   // pad_enable
                          (5u << 22)    |   // pad_interval: every 64 DWORDs
                          (1u << 25));      // pad_amount: 2 DWORDs  -> row stride 66
            g1[1] = (int)((TD & 0xFFFFu) << 16);                          // tdim0 lo16
            g1[2] = (int)(((TD >> 16) & 0xFFFFu) | ((TD & 0xFFFFu) << 16)); // tdim0 hi|tdim1 lo
            g1[3] = (int)(((TD >> 16) & 0xFFFFu) | (64u << 16));          // tdim1 hi|tile_dim0=64
            g1[4] = (int)(64u);                                           // tile_dim1=64, tile_dim2=0
            g1[5] = (int)(uint32_t)ldw;                                   // tensor_dim0_stride
            g1[6] = 0;
            g1[7] = 0;
            int32x4 z4 = {0, 0, 0, 0};
#ifdef TDM_SIX_ARGS
            int32x8 z8 = {0, 0, 0, 0, 0, 0, 0, 0};
            __builtin_amdgcn_tensor_load_to_lds(g0, g1, z4, z4, z8, 0);
#else
            __builtin_amdgcn_tensor_load_to_lds(g0, g1, z4, z4, 0);
#endif
#if __has_builtin(__builtin_amdgcn_s_wait_tensorcnt)
            __builtin_amdgcn_s_wait_tensorcnt(0);
#endif
        }
#else
        // Fallback: cooperative float4 copy with explicit padded stride.
        #pragma unroll
        for (int it = 0; it < (KC * 64) / (4 * 256); ++it) {
            const int seg = it * 256 + threadIdx.x;          // 0..1023
            const int row = seg >> 4, s4 = (seg & 15) * 4;
            *(float4*)&lw[row * LDSW + s4] =
                *(const float4*)(W + (size_t)(n0b + row) * ldw + kc + s4);
        }
#endif
        __syncthreads();

        // ---- compute: 16 k-steps x 4 n-tiles, A fragment reused 4x ----------
        #pragma unroll
        for (int k0 = 0; k0 < KC; k0 += 4) {
            v2f a = *(const v2f*)(Ap + kc + k0);
            const float* lb = &lw[(size_t)lh * LDSW + k0 + 2 * hi];
            v2f b0 = *(const v2f*)(lb);
            v2f b1 = *(const v2f*)(lb + 16 * LDSW);
            v2f b2 = *(const v2f*)(lb + 32 * LDSW);
            v2f b3 = *(const v2f*)(lb + 48 * LDSW);
            acc[0] = __builtin_amdgcn_wmma_f32_16x16x4_f32(false, a, false, b0, (short)0, acc[0], false, false);
            acc[1] = __builtin_amdgcn_wmma_f32_16x16x4_f32(false, a, false, b1, (short)0, acc[1], false, false);
            acc[2] = __builtin_amdgcn_wmma_f32_16x16x4_f32(false, a, false, b2, (short)0, acc[2], false, false);
            acc[3] = __builtin_amdgcn_wmma_f32_16x16x4_f32(false, a, false, b3, (short)0, acc[3], false, false);
        }
        __syncthreads();
    }

    // ---- fused epilogues ----------------------------------------------------
    #pragma unroll
    for (int j = 0; j < 4; ++j) {
        const int col = n0b + j * 16 + lh;
        #pragma unroll
        for (int i = 0; i < 8; i++) {
            const int row = m0 + i + hi * 8;   // C/D layout: VGPR i -> M=i (+8 for hi)
            float v = acc[j][i];
            if constexpr (EP == EP_NONE) {
                Out[(size_t)row * ldo + col] = v;
            } else if constexpr (EP == EP_SEL) {
                const float s = sigm((v + bias[col]) * gatev[col]);
                const size_t idx = (size_t)row * DD + col;
                Xout[idx] = Xin[idx] * s;
            } else if constexpr (EP == EP_IMP) {
                const float im = sigm(v + bias[col]);
                const size_t xr = (size_t)row * KX;
                const size_t o  = (size_t)row * NN + col;
                BsOut[o] = aux[xr + RR + col] * im;          // B_ssm * imp
                CsOut[o] = aux[xr + RR + NN + col] * im;     // C_ssm * imp
            } else if constexpr (EP == EP_DELTA) {
                Out[(size_t)row * ldo + col] = softpl(v + bias[col]);
            } else if constexpr (EP == EP_GATE) {
                const float g = sigm(v + bias[col]);
                const size_t idx = (size_t)row * DD + col;
                Xout[idx] = Xin[idx] * g;
            }
        }
    }
}

// ---------------------------------------------------------------------------
// Selective scan over L. Thread owns (b, d, 8 consecutive n); h & A in regs.
// y = sum_n h*c reduced with a wave32 8-lane butterfly; fuses y + x*Dskip.
// ---------------------------------------------------------------------------
__global__ __launch_bounds__(256) void scan_kernel(
    const float* __restrict__ delta, const float* __restrict__ Bs,
    const float* __restrict__ Cs,    const float* __restrict__ xg,
    const float* __restrict__ A_log, const float* __restrict__ x,
    const float* __restrict__ Dskip, float* __restrict__ z)
{
    const int t  = blockIdx.x * 256 + threadIdx.x;   // B*D*8 threads total
    const int ng = t & 7;                            // n-group within d
    const int dg = t >> 3;                           // 0 .. B*D-1
    const int b  = dg >> 10;
    const int d  = dg & (DD - 1);
    const int n0 = ng * 8;

    float a[8], h[8];
    #pragma unroll
    for (int j = 0; j < 8; j++) {
        a[j] = -__expf(A_log[(size_t)d * NN + n0 + j]);
        h[j] = 0.f;
    }
    const float dsk = Dskip[d];

    for (int l = 0; l < LL; l++) {
        const size_t rD = ((size_t)b * LL + l) * DD + d;
        const size_t rN = ((size_t)b * LL + l) * NN + n0;
        const float dt = delta[rD];
        const float xt = xg[rD];
        const float4 b0 = *(const float4*)(Bs + rN);
        const float4 b1 = *(const float4*)(Bs + rN + 4);
        const float4 c0 = *(const float4*)(Cs + rN);
        const float4 c1 = *(const float4*)(Cs + rN + 4);
        const float bt[8] = {b0.x, b0.y, b0.z, b0.w, b1.x, b1.y, b1.z, b1.w};
        const float ct[8] = {c0.x, c0.y, c0.z, c0.w, c1.x, c1.y, c1.z, c1.w};
        const float dx = dt * xt;
        float y = 0.f;
        #pragma unroll
        for (int j = 0; j < 8; j++) {
            h[j] = __expf(dt * a[j]) * h[j] + dx * bt[j];
            y = fmaf(h[j], ct[j], y);
        }
        // reduce over the 8 lanes sharing d (wave32 butterfly)
        y += __shfl_xor(y, 1, 32);
        y += __shfl_xor(y, 2, 32);
        y += __shfl_xor(y, 4, 32);
        if (ng == 0) z[rD] = y + x[rD] * dsk;
    }
}

// ---------------------------------------------------------------------------
extern "C" void kernel_launch(void* const* d_in, const int* in_sizes, int n_in,
                              void* d_out, int out_size, void* d_ws, size_t ws_size,
                              hipStream_t stream) {
    const float* x     = (const float*)d_in[0];
    const float* normw = (const float*)d_in[1];
    const float* Wx    = (const float*)d_in[2];
    const float* dtW   = (const float*)d_in[3];
    const float* dtb   = (const float*)d_in[4];
    const float* A_log = (const float*)d_in[5];
    const float* Dskip = (const float*)d_in[6];
    const float* Wout  = (const float*)d_in[7];
    const float* selW  = (const float*)d_in[8];
    const float* selb  = (const float*)d_in[9];
    const float* selg  = (const float*)d_in[10];
    const float* impW  = (const float*)d_in[11];
    const float* impb  = (const float*)d_in[12];
    const float* gateW = (const float*)d_in[13];
    const float* gateb = (const float*)d_in[14];

    float* ws = (float*)d_ws;
    const size_t SZ_BLD = (size_t)BB * LL * DD;           // 2M floats
    float* xn0   = ws;  ws += SZ_BLD;
    float* xn1   = ws;  ws += SZ_BLD;
    float* xn2   = ws;  ws += SZ_BLD;
    float* delta = ws;  ws += SZ_BLD;
    float* zbuf  = ws;  ws += SZ_BLD;
    float* xp    = ws;  ws += (size_t)BB * LL * KX;
    float* Bss   = ws;  ws += (size_t)BB * LL * NN;
    float* Css   = ws;  ws += (size_t)BB * LL * NN;

    const int M = BB * LL;                                // 2048 rows
    dim3 blk(256);

    // 1) RMSNorm
    rmsnorm_kernel<<<M, blk, 0, stream>>>(x, normw, xn0);

    // 2) xp = xn0 @ Wx^T   (2048 x 192, K=1024)
    gemm_wmma<EP_NONE><<<dim3(M / 128, KX / 64), blk, 0, stream>>>(
        xn0, DD, Wx, DD, DD, xp, KX,
        nullptr, nullptr, nullptr, nullptr, nullptr, nullptr, nullptr);

    // 3) xn1 = xn0 * sigmoid((xn0 @ selW^T + selb) * sel_gate)
    gemm_wmma<EP_SEL><<<dim3(M / 128, DD / 64), blk, 0, stream>>>(
        xn0, DD, selW, DD, DD, nullptr, 0,
        selb, selg, xn0, xn1, nullptr, nullptr, nullptr);

    // 4) imp = sigmoid(xn1 @ impW^T + impb); Bs = B_ssm*imp; Cs = C_ssm*imp
    gemm_wmma<EP_IMP><<<dim3(M / 128, NN / 64), blk, 0, stream>>>(
        xn1, DD, impW, DD, DD, nullptr, 0,
        impb, nullptr, nullptr, nullptr, xp, Bss, Css);

    // 5) delta = softplus(delta_r @ dtW^T + dtb)   (A = xp[:, :64], lda=192, K=64)
    gemm_wmma<EP_DELTA><<<dim3(M / 128, DD / 64), blk, 0, stream>>>(
        xp, KX, dtW, RR, RR, delta, DD,
        dtb, nullptr, nullptr, nullptr, nullptr, nullptr, nullptr);

    // 6) xn2 = xn1 * sigmoid(xn1 @ gateW^T + gateb)
    gemm_wmma<EP_GATE><<<dim3(M / 128, DD / 64), blk, 0, stream>>>(
        xn1, DD, gateW, DD, DD, nullptr, 0,
        gateb, nullptr, xn1, xn2, nullptr, nullptr, nullptr);

    // 7) selective scan -> z = y + x*Dskip
    scan_kernel<<<(BB * DD * 8) / 256, blk, 0, stream>>>(
        delta, Bss, Css, xn2, A_log, x, Dskip, zbuf);

    // 8) out = z @ Wout^T
    gemm_wmma<EP_NONE><<<dim3(M / 128, DD / 64), blk, 0, stream>>>(
        zbuf, DD, Wout, DD, DD, (float*)d_out, DD,
        nullptr, nullptr, nullptr, nullptr, nullptr, nullptr, nullptr);
}